// BatchCrop_5059471475189
// MI455X (gfx1250) — compile-verified
//
#include <hip/hip_runtime.h>
#include <hip/hip_bf16.h>
#include <stdint.h>

// Problem constants (fixed by the reference).
#define N_MODES   4
#define BATCH     1024
#define PH        128
#define PW        128
#define OBJ_H     2048
#define OBJ_W     2048
#define PATCH_ELEMS   (PH * PW)                 // 16384
#define PATCHES_TOTAL (BATCH * PATCH_ELEMS)     // 16,777,216 floats
#define NORM_ELEMS    (OBJ_H * OBJ_W)           // 4,194,304 floats
#define MODE_STRIDE   (BATCH * PATCH_ELEMS)     // stride between probe modes

typedef uint32_t v4u __attribute__((ext_vector_type(4)));
typedef int      v4i __attribute__((ext_vector_type(4)));
typedef int      v8i __attribute__((ext_vector_type(8)));

// ---------------------------------------------------------------------------
// Build a 2-D f32 Tensor-DMA descriptor (groups 0 and 1). Bit packing per
// CDNA5 ISA:
//   group0: [1:0]=count=1, [63:32]=lds_addr, [120:64]=global tile addr,
//           [127:126]=type=2 ("image")
//   group1: [17:16]=data_size=2 (4B), [79:48]=tensor_dim0, [111:80]=tensor_dim1,
//           [127:112]=tile_dim0, [143:128]=tile_dim1, [207:160]=dim0_stride
// Groups 2/3 zero => tile_dim2/3 unused (pure 2-D transfer).
// ---------------------------------------------------------------------------
struct TdmDesc {
    v4u g0;
    v8i g1;
};

__device__ __forceinline__ TdmDesc tdm_make_2d(uint32_t lds_off, const void* gp,
                                               uint32_t tensor_d0, uint32_t tensor_d1,
                                               uint32_t tile_d0,   uint32_t tile_d1,
                                               uint32_t stride_d0) {
    const uint64_t ga = (uint64_t)(uintptr_t)gp;
    TdmDesc d;
    d.g0.x = 1u;                                     // count=1 (valid, user mode)
    d.g0.y = lds_off;                                // lds_addr (bytes)
    d.g0.z = (uint32_t)ga;                           // global_addr[31:0]
    d.g0.w = ((uint32_t)(ga >> 32) & 0x01FFFFFFu)    // global_addr[56:32]
             | (2u << 30);                           // type = 2

    d.g1[0] = (int)(2u << 16);                                   // data_size=2 (4B)
    d.g1[1] = (int)((tensor_d0 & 0xFFFFu) << 16);                // tensor_dim0 lo16
    d.g1[2] = (int)(((tensor_d0 >> 16) & 0xFFFFu)                // tensor_dim0 hi16
                    | ((tensor_d1 & 0xFFFFu) << 16));            // tensor_dim1 lo16
    d.g1[3] = (int)(((tensor_d1 >> 16) & 0xFFFFu)                // tensor_dim1 hi16
                    | ((tile_d0 & 0xFFFFu) << 16));              // tile_dim0
    d.g1[4] = (int)(tile_d1 & 0xFFFFu);                          // tile_dim1 (dim2=0)
    d.g1[5] = (int)stride_d0;                                    // dim0_stride lo32
    d.g1[6] = 0;                                                 // stride hi bits
    d.g1[7] = 0;                                                 // dim1_stride hi32
    return d;
}

__device__ __forceinline__ void tdm_load(const TdmDesc& d) {
    const v4i gz = {0, 0, 0, 0};
#if defined(__clang_major__) && (__clang_major__ >= 23)
    const v8i gz8 = {0, 0, 0, 0, 0, 0, 0, 0};
    __builtin_amdgcn_tensor_load_to_lds(d.g0, d.g1, gz, gz, gz8, 0);
#else
    __builtin_amdgcn_tensor_load_to_lds(d.g0, d.g1, gz, gz, 0);
#endif
}

__device__ __forceinline__ void tdm_store(const TdmDesc& d) {
    const v4i gz = {0, 0, 0, 0};
#if defined(__clang_major__) && (__clang_major__ >= 23)
    const v8i gz8 = {0, 0, 0, 0, 0, 0, 0, 0};
    __builtin_amdgcn_tensor_store_from_lds(d.g0, d.g1, gz, gz, gz8, 0);
#else
    __builtin_amdgcn_tensor_store_from_lds(d.g0, d.g1, gz, gz, 0);
#endif
}

// ---------------------------------------------------------------------------
// Non-returning f32 global atomic add (avoids CAS-loop lowering of atomicAdd).
// ---------------------------------------------------------------------------
__device__ __forceinline__ void atomic_add_f32_nr(float* p, float v) {
    uint64_t ga = (uint64_t)(uintptr_t)p;
    asm volatile("global_atomic_add_f32 %0, %1, off"
                 :: "v"(ga), "v"(v)
                 : "memory");
}

// ---------------------------------------------------------------------------
// Zero-init the object_norm region of d_out (harness poisons it with 0xAA).
// ---------------------------------------------------------------------------
__global__ __launch_bounds__(256) void zero_norm_kernel(float* __restrict__ norm) {
    const int i = (blockIdx.x * 256 + threadIdx.x) * 4;
    *reinterpret_cast<float4*>(norm + i) = make_float4(0.f, 0.f, 0.f, 0.f);
}

// ---------------------------------------------------------------------------
// batch_crop: one single-wave workgroup per batch element, pure TDM DMA.
//   1) tensor_load_to_lds:   128x128 tile, row stride 2048, global -> LDS
//   2) s_wait_tensorcnt 0    (conservative; TDM is also in-order per wave)
//   3) tensor_store_from_lds: 16384x1 contiguous burst, LDS -> output patch
// The SIMD issues two descriptors and idles; the Tensor Data Mover moves
// the 64KB tile. 64KB LDS => 5 concurrent workgroups per WGP.
// ---------------------------------------------------------------------------
__global__ __launch_bounds__(32) void crop_tdm_kernel(
    const float* __restrict__ obj,
    const int*   __restrict__ pos,
    float*       __restrict__ patches)
{
    __shared__ float tile[PATCH_ELEMS];          // 64 KB of the 320 KB WGP LDS

    const int b  = blockIdx.x;
    const int p0 = pos[2 * b];
    const int p1 = pos[2 * b + 1];

    const float*   src      = obj + (size_t)p0 * OBJ_W + p1;
    float*         dst      = patches + (size_t)b * PATCH_ELEMS;
    const uint32_t lds_base = (uint32_t)(uintptr_t)(&tile[0]);

    // Strided gather: tile 128x128 from the 2048x2048 object.
    const TdmDesc dl = tdm_make_2d(lds_base, src,
                                   OBJ_W, OBJ_H,   // tensor dims (OOB never hit)
                                   PW, PH,         // tile dims
                                   OBJ_W);         // row stride
    tdm_load(dl);
    __builtin_amdgcn_s_wait_tensorcnt(0);          // tile resident in LDS

    // Contiguous drain: one 16384x1 burst into the output patch.
    const TdmDesc ds = tdm_make_2d(lds_base, dst,
                                   PATCH_ELEMS, 1, // tensor dims
                                   PATCH_ELEMS, 1, // tile dims
                                   PATCH_ELEMS);   // stride (single row)
    tdm_store(ds);
    __builtin_amdgcn_s_wait_tensorcnt(0);          // drain before endpgm
}

// ---------------------------------------------------------------------------
// overlap_intensity: one workgroup per batch element.
// patch_norm = sum over 4 modes of waves^2, scatter-added into object_norm.
// float4 (b128) coalesced reads of each mode; rows are 128 wide so a float4
// never straddles a row; per-element non-returning f32 atomics.
// ---------------------------------------------------------------------------
__global__ __launch_bounds__(256) void norm_scatter_kernel(
    const float* __restrict__ waves,
    const int*   __restrict__ pos,
    float*       __restrict__ norm)
{
    const int b   = blockIdx.x;
    const int tid = threadIdx.x;
    const int p0  = pos[2 * b];
    const int p1  = pos[2 * b + 1];

    const float* wb    = waves + (size_t)b * PATCH_ELEMS;
    float*       nbase = norm + (size_t)p0 * OBJ_W + p1;

    for (int i = tid * 4; i < PATCH_ELEMS; i += 256 * 4) {
        const int row = i >> 7;
        const int col = i & (PW - 1);

        float4 acc = make_float4(0.f, 0.f, 0.f, 0.f);
#pragma unroll
        for (int m = 0; m < N_MODES; ++m) {
            const float4 v =
                *reinterpret_cast<const float4*>(wb + (size_t)m * MODE_STRIDE + i);
            acc.x += v.x * v.x;
            acc.y += v.y * v.y;
            acc.z += v.z * v.z;
            acc.w += v.w * v.w;
        }

        float* d = nbase + (size_t)row * OBJ_W + col;
        atomic_add_f32_nr(d + 0, acc.x);
        atomic_add_f32_nr(d + 1, acc.y);
        atomic_add_f32_nr(d + 2, acc.z);
        atomic_add_f32_nr(d + 3, acc.w);
    }
}

// ---------------------------------------------------------------------------
// d_in[0] = obj   [1, 2048, 2048] f32
// d_in[1] = waves [4, 1024, 128, 128] f32
// d_in[2] = pos   [1024, 2] i32
// d_out   = patches [1024,1,128,128] f32  ++  object_norm [2048,2048] f32
// ---------------------------------------------------------------------------
extern "C" void kernel_launch(void* const* d_in, const int* in_sizes, int n_in,
                              void* d_out, int out_size, void* d_ws, size_t ws_size,
                              hipStream_t stream) {
    (void)in_sizes; (void)n_in; (void)out_size; (void)d_ws; (void)ws_size;

    const float* obj   = (const float*)d_in[0];
    const float* waves = (const float*)d_in[1];
    const int*   pos   = (const int*)d_in[2];

    float* patches = (float*)d_out;
    float* norm    = (float*)d_out + (size_t)PATCHES_TOTAL;

    // norm must be zeroed before the scatter kernel's atomics (same stream =>
    // ordered). crop is independent and can run in between.
    zero_norm_kernel<<<NORM_ELEMS / (256 * 4), 256, 0, stream>>>(norm);
    crop_tdm_kernel<<<BATCH, 32, 0, stream>>>(obj, pos, patches);
    norm_scatter_kernel<<<BATCH, 256, 0, stream>>>(waves, pos, norm);
}